// HaarDWT_37452114821793
// MI455X (gfx1250) — compile-verified
//
#include <hip/hip_runtime.h>
#include <hip/hip_bf16.h>

typedef __attribute__((ext_vector_type(2))) float v2f;
typedef __attribute__((ext_vector_type(8))) float v8f;

#define N_IMG      256            // 8*32 planes
#define IN_W       512
#define IN_PLANE   (512*512)      // 262144
#define OUT_W      256
#define OUT_H      256
#define OUT_PLANE  (OUT_W*OUT_H)  // 65536
#define PLANE_TOT  (N_IMG*OUT_PLANE) // 16,777,216 elements per output plane

// Haar level-1 DWT, one pass, bandwidth-roofline design (~512 MB total traffic).
// 4x4 Haar butterfly executed on the matrix pipe via v_wmma_f32_16x16x4_f32.
__global__ __launch_bounds__(256) void
HaarDWT_wmma_kernel(const float* __restrict__ x, float* __restrict__ out)
{
    const int lane      = threadIdx.x & 31;
    const int waveInBlk = threadIdx.x >> 5;
    const unsigned w    = blockIdx.x * 8u + (unsigned)waveInBlk; // global wave id
    const unsigned xc   = w & 7u;          // 32-pixel chunk along output row
    const unsigned y    = (w >> 3) & 255u; // output row
    const unsigned img  = w >> 11;         // which of the 256 planes

    // ---------------- Constant A matrix (16x4), 2 VGPRs per lane ----------------
    // ISA layout: lanes 0-15 hold rows M=lane, K={0,1}; lanes 16-31 rows M=lane-16, K={2,3}.
    // Rows 0-3 = 0.25*M (LL,LH,HL,HH); rows 8-11 replicate rows 0-3 so the upper
    // half of D duplicates the useful outputs; rows 4-7 / 12-15 are zero.
    const int m    = lane & 15;
    const int kb   = (lane >> 4) << 1;        // K base: 0 or 2
    const int mrow = m & 3;                   // row within the 4x4 M
    const bool valid = (m & 4) == 0;          // rows {0..3, 8..11}
    v2f A;
#pragma unroll
    for (int j = 0; j < 2; ++j) {
        const int k  = kb + j;
        const int kr = (k >> 1) & 1;          // input row parity  (a,b vs c,d)
        const int kc = k & 1;                 // input col parity  (a,c vs b,d)
        // sign exponent: LL:+ ; LH: -(row even); HL: -(col even); HH: (-1)^(kr^kc)
        const int e  = ((mrow & 1) * (kr + 1) + ((mrow >> 1) & 1) * (kc + 1)) & 1;
        A[j] = valid ? (e ? -0.25f : 0.25f) : 0.0f;
    }

    // ---------------- Load B matrices (two 4x16 tiles) ----------------
    // Lane l reads input row 2y+(l>=16), pixel p = xc*32 + (l&15) (+16 for tile 1):
    // float2 = (even col, odd col) -> (B0,B1) = (K0|K2, K1|K3). 128B contiguous
    // per half-wave, every input byte touched exactly once -> non-temporal.
    const unsigned row  = 2u * y + (unsigned)((lane >> 4) & 1);
    const v2f* inrow = (const v2f*)(x + (size_t)img * IN_PLANE + (size_t)row * IN_W);
    const unsigned p0 = xc * 32u + (unsigned)(lane & 15);
    v2f b_t0 = __builtin_nontemporal_load(&inrow[p0]);
    v2f b_t1 = __builtin_nontemporal_load(&inrow[p0 + 16u]);

    // ---------------- Matrix butterfly on the WMMA pipe ----------------
    v8f cz = {};
    v8f d0 = __builtin_amdgcn_wmma_f32_16x16x4_f32(false, A, false, b_t0,
                                                   (short)0, cz, false, false);
    v8f d1 = __builtin_amdgcn_wmma_f32_16x16x4_f32(false, A, false, b_t1,
                                                   (short)0, cz, false, false);

    // ---------------- Store: merge tiles into full 128B coalesced rows ----------
    // D VGPR v: lanes 0-15 = plane v (tile0 pixels), lanes 16-31 = replicated
    // plane v (rows 8-11) of tile1 -> select gives 32 consecutive output pixels.
    const size_t outOff = (size_t)img * OUT_PLANE + (size_t)y * OUT_W
                        + (size_t)xc * 32u + (size_t)lane;
    const bool lo = lane < 16;
#pragma unroll
    for (int v = 0; v < 4; ++v) {
        float val = lo ? d0[v] : d1[v];
        __builtin_nontemporal_store(val, out + (size_t)v * PLANE_TOT + outOff);
    }
}

extern "C" void kernel_launch(void* const* d_in, const int* in_sizes, int n_in,
                              void* d_out, int out_size, void* d_ws, size_t ws_size,
                              hipStream_t stream) {
    (void)in_sizes; (void)n_in; (void)out_size; (void)d_ws; (void)ws_size;
    const float* x = (const float*)d_in[0];
    float* out = (float*)d_out;
    // 256 planes * 256 rows * 8 chunks = 524,288 wave-tasks; 8 waves per 256-thread block.
    dim3 grid(65536), block(256);
    HaarDWT_wmma_kernel<<<grid, block, 0, stream>>>(x, out);
}